// Reconstruction_15831249453217
// MI455X (gfx1250) — compile-verified
//
#include <hip/hip_runtime.h>
#include <math.h>

// ---------------------------------------------------------------------------
// DRR forward: out[b,r] = mean_p( trilerp( sigmoid(raw-6), src + a_p*(tgt-src) ) )
//                         * || (tgt-src) * spacing ||
// Volume is 256^3 fp32 (64 MB) -> fully L2-resident on MI455X (192 MB L2).
// Strategy:
//   K1: precompute sigmoid volume into d_ws (one pass, v_exp_f32)
//   K2: 16 rays per wave-group; lane l of each wave owns ray l%16 and computes
//       2 trilinear samples per step, fed into V_WMMA_F32_16X16X4_F32 with
//       B = ones and chained C accumulation -> D row m = sample-sum of ray m.
//       4 waves split the 400 samples; partial D tiles combined via LDS.
//   Branch-free inner loop (value selects, not EXEC masks) so EXEC stays
//   all-ones for WMMA; WGP-scope prefetch walks ahead along the ray.
// ---------------------------------------------------------------------------

#define SHIFT_C 6.0f
#define DIM 256          // D = H = W = 256 (setup_inputs)
#define DMAXF 255.0f
#define DMAXI 255

typedef __attribute__((ext_vector_type(2))) float v2f;
typedef __attribute__((ext_vector_type(8))) float v8f;

__device__ __forceinline__ float sigm(float x) {
    // sigmoid(x - 6) = 1 / (1 + e^(6-x))
    return 1.0f / (1.0f + __expf(SHIFT_C - x));
}

// ---------------- Kernel 1: sigmoid precompute ------------------------------
__global__ void __launch_bounds__(256)
sigmoid_vol_kernel(const float* __restrict__ raw, float* __restrict__ dst, int n4) {
    int i = blockIdx.x * blockDim.x + threadIdx.x;
    if (i >= n4) return;
    float4 v = reinterpret_cast<const float4*>(raw)[i];
    float4 r;
    r.x = sigm(v.x); r.y = sigm(v.y); r.z = sigm(v.z); r.w = sigm(v.w);
    reinterpret_cast<float4*>(dst)[i] = r;
}

// ---------------- trilinear sample (alpha pre-clamped to [0,1]) -------------
template<bool PRE>
__device__ __forceinline__ float trilerp_sample(const float* __restrict__ vol,
                                                float s0, float s1, float s2,
                                                float d0, float d1, float d2,
                                                float al) {
    // position (coord0 = z-dim, coord1 = y, coord2 = x), border-clamped
    float p0 = fminf(fmaxf(fmaf(al, d0, s0), 0.0f), DMAXF);
    float p1 = fminf(fmaxf(fmaf(al, d1, s1), 0.0f), DMAXF);
    float p2 = fminf(fmaxf(fmaf(al, d2, s2), 0.0f), DMAXF);
    float q0 = floorf(p0), q1 = floorf(p1), q2 = floorf(p2);
    float f0 = p0 - q0,   f1 = p1 - q1,   f2 = p2 - q2;
    int i0 = (int)q0, i1 = (int)q1, i2 = (int)q2;
    int j0 = min(i0 + 1, DMAXI), j1 = min(i1 + 1, DMAXI), j2 = min(i2 + 1, DMAXI);
    // linear index: (z<<16) | (y<<8) | x   for a 256^3 volume
    int b00 = (i0 << 16) + (i1 << 8);
    int b01 = (i0 << 16) + (j1 << 8);
    int b10 = (j0 << 16) + (i1 << 8);
    int b11 = (j0 << 16) + (j1 << 8);
    float c000 = vol[b00 + i2], c001 = vol[b00 + j2];
    float c010 = vol[b01 + i2], c011 = vol[b01 + j2];
    float c100 = vol[b10 + i2], c101 = vol[b10 + j2];
    float c110 = vol[b11 + i2], c111 = vol[b11 + j2];
    if (!PRE) {
        c000 = sigm(c000); c001 = sigm(c001); c010 = sigm(c010); c011 = sigm(c011);
        c100 = sigm(c100); c101 = sigm(c101); c110 = sigm(c110); c111 = sigm(c111);
    }
    float c00 = fmaf(f2, c001 - c000, c000);
    float c01 = fmaf(f2, c011 - c010, c010);
    float c10 = fmaf(f2, c101 - c100, c100);
    float c11 = fmaf(f2, c111 - c110, c110);
    float c0  = fmaf(f1, c01 - c00, c00);
    float c1  = fmaf(f1, c11 - c10, c10);
    return fmaf(f0, c1 - c0, c0);
}

// ---------------- Kernel 2: ray march + WMMA reduction ----------------------
template<bool PRE>
__global__ void __launch_bounds__(128)
drr_kernel(const float* __restrict__ vol,
           const float* __restrict__ spacing,
           const float* __restrict__ src,
           const float* __restrict__ tgt,
           const int*   __restrict__ n_points_p,
           float* __restrict__ out,
           int nRays) {
    const int tid  = threadIdx.x;
    const int lane = tid & 31;          // wave32
    const int wave = tid >> 5;          // 0..3 (sample-chunk splitter)
    const int m    = lane & 15;         // ray-in-group == A-matrix row (ISA layout)
    int rayIdx = blockIdx.x * 16 + m;
    int ray = min(rayIdx, nRays - 1);   // clamp for safe loads on ragged tail

    const float s0 = src[ray * 3 + 0], s1 = src[ray * 3 + 1], s2 = src[ray * 3 + 2];
    const float t0 = tgt[ray * 3 + 0], t1 = tgt[ray * 3 + 1], t2 = tgt[ray * 3 + 2];
    const float d0 = t0 - s0, d1 = t1 - s1, d2 = t2 - s2;

    const int   P      = *n_points_p;                 // 400
    const float invPm1 = 1.0f / (float)(P - 1);
    const float inc16  = 16.0f * invPm1;              // prefetch look-ahead step
    const int   nChunk = (P + 3) >> 2;                // K-chunks of 4 samples
    const int   koff   = (lane >> 4) << 1;            // lanes 0-15: K=0,1 ; 16-31: K=2,3

    v2f a;
    v2f bOnes; bOnes.x = 1.0f; bOnes.y = 1.0f;        // B = all-ones 4x16
    v8f acc = {0.f,0.f,0.f,0.f,0.f,0.f,0.f,0.f};

    #pragma unroll 2
    for (int ci = wave; ci < nChunk; ci += 4) {
        const int   p0  = (ci << 2) + koff;
        const float al0 = fminf((float)p0 * invPm1, 1.0f);
        const float al1 = fminf(al0 + invPm1, 1.0f);

        // WGP-scope prefetch of this wave's NEXT chunk's base corner
        {
            float aln = al0 + inc16;
            float q0 = fminf(fmaxf(fmaf(aln, d0, s0), 0.0f), DMAXF);
            float q1 = fminf(fmaxf(fmaf(aln, d1, s1), 0.0f), DMAXF);
            float q2 = fminf(fmaxf(fmaf(aln, d2, s2), 0.0f), DMAXF);
            int bz = (int)q0, by = (int)q1, bx = (int)q2;
            __builtin_prefetch(vol + ((bz << 16) + (by << 8) + bx), 0, 3);
        }

        // branch-free: compute unconditionally (alpha clamped keeps loads
        // in-bounds), zero tail contributions with value selects so EXEC
        // stays all-ones for WMMA
        float a0 = trilerp_sample<PRE>(vol, s0, s1, s2, d0, d1, d2, al0);
        float a1 = trilerp_sample<PRE>(vol, s0, s1, s2, d0, d1, d2, al1);
        a.x = (p0     < P) ? a0 : 0.0f;
        a.y = (p0 + 1 < P) ? a1 : 0.0f;

        // D[m,n] = sum_k A[m,k]*1 + C[m,n]  -> accumulates per-ray sample sum
        acc = __builtin_amdgcn_wmma_f32_16x16x4_f32(
                  false, a, false, bOnes, (short)0, acc, false, false);
    }

    // combine the 4 waves' partial D tiles; each D column is identical, read col 0
    __shared__ float tile[4][16][16];
    const int mrow = (lane >> 4) << 3;    // lanes 0-15 hold rows 0-7, 16-31 rows 8-15
    #pragma unroll
    for (int r = 0; r < 8; ++r)
        tile[wave][mrow + r][m] = acc[r];
    __syncthreads();

    if (tid < 16 && rayIdx < nRays) {
        float sum = tile[0][tid][0] + tile[1][tid][0]
                  + tile[2][tid][0] + tile[3][tid][0];
        // ray length: this thread's (s,t) already belong to ray m == tid
        float e0 = d0 * spacing[0], e1 = d1 * spacing[1], e2 = d2 * spacing[2];
        float rl = sqrtf(fmaf(e0, e0, fmaf(e1, e1, e2 * e2)));
        out[rayIdx] = sum * (1.0f / (float)P) * rl;
    }
}

// ---------------------------------------------------------------------------
extern "C" void kernel_launch(void* const* d_in, const int* in_sizes, int n_in,
                              void* d_out, int out_size, void* d_ws, size_t ws_size,
                              hipStream_t stream) {
    const float* raw     = (const float*)d_in[0];   // [256,256,256]
    const float* spacing = (const float*)d_in[1];   // [3]
    const float* src     = (const float*)d_in[2];   // [B,R,3]
    const float* tgt     = (const float*)d_in[3];   // [B,R,3]
    const int*   npts    = (const int*)  d_in[4];   // scalar (400)
    float*       out     = (float*)d_out;

    const int volElems = in_sizes[0];               // 16,777,216
    const int nRays    = in_sizes[2] / 3;           // 65,536

    dim3 blk(128);
    dim3 grd((nRays + 15) / 16);

    if (ws_size >= (size_t)volElems * sizeof(float)) {
        float* volSig = (float*)d_ws;
        int n4 = volElems / 4;
        sigmoid_vol_kernel<<<dim3((n4 + 255) / 256), dim3(256), 0, stream>>>(raw, volSig, n4);
        drr_kernel<true><<<grd, blk, 0, stream>>>(volSig, spacing, src, tgt, npts, out, nRays);
    } else {
        drr_kernel<false><<<grd, blk, 0, stream>>>(raw, spacing, src, tgt, npts, out, nRays);
    }
}